// Wav2Vec2GumbelVectorQuantizer_8899172238060
// MI455X (gfx1250) — compile-verified
//
#include <hip/hip_runtime.h>
#include <hip/hip_bf16.h>
#include <math.h>

// ---------------------------------------------------------------------------
// Wav2Vec2 Gumbel Vector Quantizer, fused for gfx1250 (MI455X).
//   B=32, S=2048, D=512, G=2, K=320, GK=640, dpg=128, N=B*S=65536
// Pipeline:
//   k_main : fp32 WMMA GEMM (logits) -> LDS -> softmax stats + gumbel argmax
//            -> codevector gather to d_out, per-block marginal partials to ws
//   k_final: deterministic reduction of partials -> perplexity at d_out[last]
// ---------------------------------------------------------------------------

typedef float v2f __attribute__((ext_vector_type(2)));
typedef float v8f __attribute__((ext_vector_type(8)));

#define DIM     512
#define GK      640
#define KVARS   320
#define NGROUP  2
#define DPG     128
#define MTILE   32
#define LSTRIDE 644   // 640 + pad: breaks the 2-way bank conflict on the D-frag store
#define TAU_EPS 1e-7f

__global__ __launch_bounds__(256) void k_main(
    const float* __restrict__ hid,        // (N, 512)
    const unsigned char* __restrict__ msk,// (N,) bool
    const float* __restrict__ gum,        // (N*2, 320)
    const float* __restrict__ Wm,         // (640, 512)
    const float* __restrict__ bias,       // (640,)
    const float* __restrict__ cv,         // (640, 128)
    float* __restrict__ out,              // (N, 256)
    float* __restrict__ wsPartial)        // (gridDim.x, 640)
{
    extern __shared__ float smem[];
    float* sLog  = smem;                    // MTILE * LSTRIDE
    float* sMarg = smem + MTILE * LSTRIDE;  // 8 * 320

    const int tid    = threadIdx.x;
    const int wave   = tid >> 5;
    const int lane   = tid & 31;
    const int laneLo = lane & 15;
    const int laneHi = lane >> 4;           // 0 | 1
    const int rowSub = wave & 1;            // 2 row subtiles of 16
    const int strip  = wave >> 1;           // 4 col strips of 160 (=10 subtiles)
    const int colBase = strip * 160;
    const size_t blockRow = (size_t)blockIdx.x * MTILE;

    // ---------------- Phase 1: logits tile via V_WMMA_F32_16X16X4_F32 -------
    // f32 A frag (16x4): lanes 0-15 rows, VGPR0/1 = K0/K1 (lo half) K2/K3 (hi)
    //   -> per lane: one b64 load of hid[row][k0 + 2*laneHi .. +1]
    // f32 B frag (4x16): lanes 0-15 cols, same K split
    //   -> per lane: one b64 load of W[col][k0 + 2*laneHi .. +1]
    v8f acc[10];
#pragma unroll
    for (int c = 0; c < 10; ++c) acc[c] = (v8f){0.f,0.f,0.f,0.f,0.f,0.f,0.f,0.f};

    const size_t rowA = blockRow + (size_t)(rowSub * 16 + laneLo);
    const float* aBase = hid + rowA * DIM + 2 * laneHi;

    for (int k0 = 0; k0 < DIM; k0 += 4) {
        v2f a = *(const v2f*)(aBase + k0);
#pragma unroll
        for (int c = 0; c < 10; ++c) {
            const int col = colBase + c * 16 + laneLo;
            v2f bf = *(const v2f*)(Wm + (size_t)col * DIM + k0 + 2 * laneHi);
            acc[c] = __builtin_amdgcn_wmma_f32_16x16x4_f32(
                         false, a, false, bf, (short)0, acc[c], false, false);
        }
    }

    // D frag: element v -> row = rowSub*16 + laneHi*8 + v, col = lane%16 slot
#pragma unroll
    for (int c = 0; c < 10; ++c) {
        const int col = colBase + c * 16 + laneLo;
        const float bv = bias[col];
#pragma unroll
        for (int v = 0; v < 8; ++v) {
            const int rloc = rowSub * 16 + laneHi * 8 + v;
            sLog[rloc * LSTRIDE + col] = acc[c][v] + bv;
        }
    }
    __syncthreads();

    // ---------------- Phase 2: softmax / gumbel-argmax / gather -------------
    const int g     = wave & 1;
    const int rbase = wave >> 1;

    float marg[10];
#pragma unroll
    for (int j = 0; j < 10; ++j) marg[j] = 0.f;

    for (int i = 0; i < 8; ++i) {
        const int rloc = rbase + 4 * i;
        const size_t n = blockRow + rloc;

        float lv[10];
#pragma unroll
        for (int j = 0; j < 10; ++j)
            lv[j] = sLog[rloc * LSTRIDE + g * KVARS + j * 32 + lane];

        // plain softmax (for perplexity marginal)
        float mx = lv[0];
#pragma unroll
        for (int j = 1; j < 10; ++j) mx = fmaxf(mx, lv[j]);
        for (int m = 16; m >= 1; m >>= 1) mx = fmaxf(mx, __shfl_xor(mx, m, 32));

        float ex[10];
        float s = 0.f;
#pragma unroll
        for (int j = 0; j < 10; ++j) { ex[j] = __expf(lv[j] - mx); s += ex[j]; }
        for (int m = 16; m >= 1; m >>= 1) s += __shfl_xor(s, m, 32);
        const float inv = __frcp_rn(s);
        const float mk  = (msk[n] != 0) ? 1.f : 0.f;
#pragma unroll
        for (int j = 0; j < 10; ++j) marg[j] = fmaf(ex[j] * inv, mk, marg[j]);

        // hard path: probs == one_hot(argmax(logit + gumbel))  (tau monotonic)
        const float* gp = gum + ((size_t)n * NGROUP + g) * KVARS;
        float bvv = -INFINITY; int bi = KVARS;
#pragma unroll
        for (int j = 0; j < 10; ++j) {
            const int k = j * 32 + lane;
            const float vv = lv[j] + gp[k];
            if (vv > bvv) { bvv = vv; bi = k; }
        }
        for (int m = 16; m >= 1; m >>= 1) {
            const float ov = __shfl_xor(bvv, m, 32);
            const int   oi = __shfl_xor(bi,  m, 32);
            if (ov > bvv || (ov == bvv && oi < bi)) { bvv = ov; bi = oi; }
        }

        // gather selected codevector (128 floats, float4/lane, coalesced)
        const float4 cvv = *(const float4*)(cv + ((size_t)(g * KVARS + bi)) * DPG + lane * 4);
        *(float4*)(out + n * (NGROUP * DPG) + g * DPG + lane * 4) = cvv;
    }

    // deterministic per-block marginal partial: fixed-order 4-wave combine
#pragma unroll
    for (int j = 0; j < 10; ++j) sMarg[wave * KVARS + j * 32 + lane] = marg[j];
    __syncthreads();

    for (int t = tid; t < GK; t += 256) {
        const int gg = t / KVARS, k = t % KVARS;
        const float sum = sMarg[(gg + 0) * KVARS + k] + sMarg[(gg + 2) * KVARS + k]
                        + sMarg[(gg + 4) * KVARS + k] + sMarg[(gg + 6) * KVARS + k];
        wsPartial[(size_t)blockIdx.x * GK + t] = sum;
    }
}

__global__ __launch_bounds__(1024) void k_final(
    const float* __restrict__ wsPartial,
    const unsigned char* __restrict__ msk,
    float* __restrict__ outPerp,
    int nBlocks, int nMask)
{
    __shared__ float sEnt[GK];
    __shared__ int   sCnt[1024];
    const int tid = threadIdx.x;

    int c = 0;
    for (int i = tid; i < nMask; i += 1024) c += (msk[i] != 0) ? 1 : 0;
    sCnt[tid] = c;
    __syncthreads();
    for (int s = 512; s > 0; s >>= 1) {
        if (tid < s) sCnt[tid] += sCnt[tid + s];
        __syncthreads();
    }
    const float invCnt = 1.0f / (float)sCnt[0];

    if (tid < GK) {
        float sum = 0.f;
        for (int b = 0; b < nBlocks; ++b)            // fixed order: deterministic
            sum += wsPartial[(size_t)b * GK + tid];
        const float m = sum * invCnt;
        sEnt[tid] = m * __logf(m + TAU_EPS);
    }
    __syncthreads();
    if (tid == 0) {
        float e0 = 0.f, e1 = 0.f;
        for (int k = 0; k < KVARS; ++k) { e0 += sEnt[k]; e1 += sEnt[KVARS + k]; }
        *outPerp = __expf(-e0) + __expf(-e1);
    }
}

extern "C" void kernel_launch(void* const* d_in, const int* in_sizes, int n_in,
                              void* d_out, int out_size, void* d_ws, size_t ws_size,
                              hipStream_t stream)
{
    const float*         hid  = (const float*)d_in[0];
    const unsigned char* msk  = (const unsigned char*)d_in[1];   // jnp bool -> 1 byte
    const float*         gum  = (const float*)d_in[2];
    const float*         Wm   = (const float*)d_in[3];
    const float*         bias = (const float*)d_in[4];
    const float*         cv   = (const float*)d_in[5];
    float*               out  = (float*)d_out;
    float*               wsP  = (float*)d_ws;                    // nBlocks*640 floats

    const int N       = in_sizes[1];        // B*S = 65536
    const int nBlocks = N / MTILE;          // 2048
    const size_t shmem = (size_t)(MTILE * LSTRIDE + 8 * KVARS) * sizeof(float);

    k_main<<<nBlocks, 256, shmem, stream>>>(hid, msk, gum, Wm, bias, cv, out, wsP);
    k_final<<<1, 1024, 0, stream>>>(wsP, msk, out + (size_t)out_size - 1, nBlocks, N);
}